// GAT_4148938408329
// MI455X (gfx1250) — compile-verified
//
#include <hip/hip_runtime.h>
#include <hip/hip_bf16.h>
#include <math.h>

#define N_NODES 50000
#define N_EDGES 1600000
#define IN_C    128
#define HEADS   4
#define OUT_C   16
#define HC      64   // HEADS*OUT_C

typedef __attribute__((ext_vector_type(2))) float v2f;
typedef __attribute__((ext_vector_type(8))) float v8f;

// Monotonic unsigned encoding of f32 so segment-max can use HW atomic_max_u32.
__device__ __forceinline__ unsigned enc_f32(float f) {
  unsigned u = __float_as_uint(f);
  return (u & 0x80000000u) ? ~u : (u | 0x80000000u);
}
__device__ __forceinline__ float dec_f32(unsigned e) {
  unsigned u = (e & 0x80000000u) ? (e ^ 0x80000000u) : ~e;
  return __uint_as_float(u);
}
#define ENC_NEG_INF 0x007FFFFFu  // enc(-inf)

// ---------------------------------------------------------------------------
// K1: xw = x @ W via V_WMMA_F32_16X16X4_F32.
// Block = 128 threads = 4 waves; wave w computes the 16x16 tile for head w.
// Block b covers node rows [b*16, b*16+16). 50000/16 = 3125 exactly.
// A frag (16x4 f32): lanes 0-15 hold K=0,1; lanes 16-31 hold K=2,3 (per ISA).
// B frag (4x16 f32): lanes 0-15 -> N=lane,K=0/1; lanes 16-31 -> N=lane-16,K=2/3.
// C/D (16x16 f32): vgpr j, lanes 0-15 -> M=j; lanes 16-31 -> M=j+8.
// ---------------------------------------------------------------------------
__global__ __launch_bounds__(128) void gat_gemm_wmma(
    const float* __restrict__ x, const float* __restrict__ W,
    float* __restrict__ xw) {
  const int wave = threadIdx.x >> 5;
  const int lane = threadIdx.x & 31;
  const int row0 = blockIdx.x * 16;
  const int col0 = wave * OUT_C;       // head == wave
  const int lrow = lane & 15;
  const int hi   = lane >> 4;          // 0 or 1
  const int koff = hi * 2;             // K offset of this half-wave

  const float* __restrict__ xr = x + (size_t)(row0 + lrow) * IN_C;
  const float* __restrict__ wc = W + col0 + lrow;

  v8f c = {};
#pragma unroll
  for (int k = 0; k < IN_C; k += 4) {
    v2f a = *(const v2f*)(xr + k + koff);           // two consecutive K of row
    v2f b;
    b.x = wc[(size_t)(k + koff)     * HC];
    b.y = wc[(size_t)(k + koff + 1) * HC];
    c = __builtin_amdgcn_wmma_f32_16x16x4_f32(
        /*neg_a=*/false, a, /*neg_b=*/false, b,
        /*c_mod=*/(short)0, c, /*reuse_a=*/false, /*reuse_b=*/false);
  }

  float* __restrict__ o = xw + (size_t)(row0 + hi * 8) * HC + col0 + lrow;
#pragma unroll
  for (int j = 0; j < 8; ++j) o[(size_t)j * HC] = c[j];
}

// ---------------------------------------------------------------------------
// K2: zero agg/denom, init segment-max to enc(-inf).
// ---------------------------------------------------------------------------
__global__ void gat_init(float* __restrict__ agg, float* __restrict__ denom,
                         unsigned* __restrict__ amax) {
  int t = blockIdx.x * blockDim.x + threadIdx.x;
  if (t < N_NODES * HC) agg[t] = 0.0f;
  if (t < N_NODES * HEADS) { denom[t] = 0.0f; amax[t] = ENC_NEG_INF; }
}

// ---------------------------------------------------------------------------
// K3: per-node attention logits a_src/a_dst: [N, H].
// ---------------------------------------------------------------------------
__global__ void gat_logits(const float* __restrict__ xw,
                           const float* __restrict__ att_src,
                           const float* __restrict__ att_dst,
                           float* __restrict__ a_src, float* __restrict__ a_dst) {
  int t = blockIdx.x * blockDim.x + threadIdx.x;
  if (t >= N_NODES * HEADS) return;
  int n = t >> 2, h = t & 3;
  const float* v = xw + (size_t)n * HC + h * OUT_C;
  float s = 0.0f, d = 0.0f;
#pragma unroll
  for (int c = 0; c < OUT_C; ++c) {
    float xv = v[c];
    s = fmaf(xv, att_src[h * OUT_C + c], s);
    d = fmaf(xv, att_dst[h * OUT_C + c], d);
  }
  a_src[t] = s;
  a_dst[t] = d;
}

// ---------------------------------------------------------------------------
// K4: per-edge raw score (leaky relu 0.2) + segment max over dst.
// ---------------------------------------------------------------------------
__global__ void gat_edge_alpha_max(const int* __restrict__ ei,
                                   const float* __restrict__ a_src,
                                   const float* __restrict__ a_dst,
                                   float* __restrict__ alpha,
                                   unsigned* __restrict__ amax) {
  int t = blockIdx.x * blockDim.x + threadIdx.x;
  if (t >= N_EDGES * HEADS) return;
  int e = t >> 2, h = t & 3;
  int s = ei[e];
  int d = ei[N_EDGES + e];
  float al = a_src[s * HEADS + h] + a_dst[d * HEADS + h];
  al = (al > 0.0f) ? al : 0.2f * al;
  alpha[t] = al;
  atomicMax(&amax[d * HEADS + h], enc_f32(al));
}

// ---------------------------------------------------------------------------
// K5: ex = exp(alpha - max) (overwrites alpha buffer) + segment sum.
// ---------------------------------------------------------------------------
__global__ void gat_edge_expsum(const int* __restrict__ ei,
                                const unsigned* __restrict__ amax,
                                float* __restrict__ alpha,
                                float* __restrict__ denom) {
  int t = blockIdx.x * blockDim.x + threadIdx.x;
  if (t >= N_EDGES * HEADS) return;
  int e = t >> 2, h = t & 3;
  int d = ei[N_EDGES + e];
  float m = dec_f32(amax[d * HEADS + h]);
  float v = __expf(alpha[t] - m);
  alpha[t] = v;
  atomicAdd(&denom[d * HEADS + h], v);
}

// ---------------------------------------------------------------------------
// K6: weighted scatter-add aggregation. 16 lanes per edge, float4 per lane.
// xw (12.8MB) and agg (12.8MB) stay L2-resident (192MB L2).
// ---------------------------------------------------------------------------
__global__ void gat_aggregate(const int* __restrict__ ei,
                              const float* __restrict__ xw,
                              const float* __restrict__ exv,
                              const float* __restrict__ denom,
                              float* __restrict__ agg) {
  int t = blockIdx.x * blockDim.x + threadIdx.x;
  int e = t >> 4;
  if (e >= N_EDGES) return;
  int q = t & 15;          // channel group: channels [q*4, q*4+4)
  int h = q >> 2;          // head of this channel group
  int s = ei[e];
  int d = ei[N_EDGES + e];
  float w = exv[e * HEADS + h] / (denom[d * HEADS + h] + 1e-16f);
  float4 m = *(const float4*)(xw + (size_t)s * HC + q * 4);
  float* o = agg + (size_t)d * HC + q * 4;
  atomicAdd(o + 0, m.x * w);
  atomicAdd(o + 1, m.y * w);
  atomicAdd(o + 2, m.z * w);
  atomicAdd(o + 3, m.w * w);
}

// ---------------------------------------------------------------------------
// K7: out = elu(agg + bias); y = out @ W_out + b_out. One wave per node,
// lane owns channels {lane, lane+32}, shuffle-reduce (wave32).
// ---------------------------------------------------------------------------
__global__ void gat_finalize(const float* __restrict__ agg,
                             const float* __restrict__ bias,
                             const float* __restrict__ W_out,
                             const float* __restrict__ b_out,
                             float* __restrict__ y) {
  int t = blockIdx.x * blockDim.x + threadIdx.x;
  int n = t >> 5;
  int lane = t & 31;
  if (n >= N_NODES) return;
  float acc = 0.0f;
#pragma unroll
  for (int i = 0; i < 2; ++i) {
    int c = lane + i * 32;
    float v = agg[(size_t)n * HC + c] + bias[c];
    v = (v > 0.0f) ? v : (__expf(v) - 1.0f);   // ELU(alpha=1)
    acc = fmaf(v, W_out[c], acc);
  }
#pragma unroll
  for (int off = 16; off > 0; off >>= 1) acc += __shfl_down(acc, off, 32);
  if (lane == 0) y[n] = acc + b_out[0];
}

// ---------------------------------------------------------------------------
extern "C" void kernel_launch(void* const* d_in, const int* in_sizes, int n_in,
                              void* d_out, int out_size, void* d_ws, size_t ws_size,
                              hipStream_t stream) {
  const float* x       = (const float*)d_in[0];
  const int*   ei      = (const int*)d_in[1];   // [2, E]: row0=src, row1=dst
  const float* W       = (const float*)d_in[2];
  const float* att_src = (const float*)d_in[3];
  const float* att_dst = (const float*)d_in[4];
  const float* bias    = (const float*)d_in[5];
  const float* W_out   = (const float*)d_in[6];
  const float* b_out   = (const float*)d_in[7];
  float* y = (float*)d_out;

  // Workspace layout (floats): ~54.4 MB total
  float* ws    = (float*)d_ws;
  float* xw    = ws;                                   // N*64
  float* a_src = xw    + (size_t)N_NODES * HC;         // N*4
  float* a_dst = a_src + (size_t)N_NODES * HEADS;      // N*4
  float* denom = a_dst + (size_t)N_NODES * HEADS;      // N*4
  float* agg   = denom + (size_t)N_NODES * HEADS;      // N*64
  float* alpha = agg   + (size_t)N_NODES * HC;         // E*4 (alpha, then ex)
  unsigned* amax = (unsigned*)(alpha + (size_t)N_EDGES * HEADS); // N*4

  gat_gemm_wmma<<<N_NODES / 16, 128, 0, stream>>>(x, W, xw);
  gat_init<<<(N_NODES * HC + 255) / 256, 256, 0, stream>>>(agg, denom, amax);
  gat_logits<<<(N_NODES * HEADS + 255) / 256, 256, 0, stream>>>(
      xw, att_src, att_dst, a_src, a_dst);
  gat_edge_alpha_max<<<(N_EDGES * HEADS + 255) / 256, 256, 0, stream>>>(
      ei, a_src, a_dst, alpha, amax);
  gat_edge_expsum<<<(N_EDGES * HEADS + 255) / 256, 256, 0, stream>>>(
      ei, amax, alpha, denom);
  gat_aggregate<<<(N_EDGES * 16 + 255) / 256, 256, 0, stream>>>(
      ei, xw, alpha, denom, agg);
  gat_finalize<<<(N_NODES * 32 + 255) / 256, 256, 0, stream>>>(
      agg, bias, W_out, b_out, y);
}